// ListMLE_38019050504793
// MI455X (gfx1250) — compile-verified
//
#include <hip/hip_runtime.h>
#include <cmath>
#include <cstdint>

#define B_ROWS 8192
#define N_COLS 4096
#define LANES  32
#define SEG    (N_COLS / LANES)   // 128 elements per lane
#define CHLEN  32                 // 4 chains of 32 per lane

// Numerically stable logaddexp, matching jnp.logaddexp semantics
// (handles the -inf identity used for the exclusive-scan seed).
__device__ __forceinline__ float lae(float a, float b) {
    float mx = fmaxf(a, b);
    if (mx == -INFINITY) return -INFINITY;
    float mn = fminf(a, b);
    return mx + log1pf(expf(mn - mx));
}

__global__ __launch_bounds__(32) void listmle_rows(
    const float* __restrict__ outputs,
    const int*   __restrict__ labels,
    float*       __restrict__ partial)
{
    __shared__ __align__(16) float s_out[N_COLS];   // 16 KB: row of outputs
    __shared__ __align__(16) int   s_lab[N_COLS];   // 16 KB: row of labels, reused as L-buffer
    float* Lbuf = (float*)s_lab;

    const int row  = blockIdx.x;
    const int lane = threadIdx.x;

    const char* gout = (const char*)(outputs + (size_t)row * N_COLS);
    const char* glab = (const char*)(labels  + (size_t)row * N_COLS);

    // Low 32 bits of a generic shared pointer = workgroup-relative LDS byte address.
    const unsigned lds_out = (unsigned)(uintptr_t)&s_out[0];
    const unsigned lds_lab = (unsigned)(uintptr_t)&s_lab[0];

    // Async DMA: 32 chunks x 512B per buffer. GVS mode: saddr(64b SGPR) + vaddr(32b VGPR).
    // The same per-lane offset serves as both the global offset and the LDS offset.
#pragma unroll
    for (int c = 0; c < 32; ++c) {
        unsigned off = (unsigned)(lane * 16 + c * 512);
        asm volatile("global_load_async_to_lds_b128 %0, %1, %2"
                     :: "v"(lds_out + off), "v"(off), "s"(gout) : "memory");
    }
#pragma unroll
    for (int c = 0; c < 32; ++c) {
        unsigned off = (unsigned)(lane * 16 + c * 512);
        asm volatile("global_load_async_to_lds_b128 %0, %1, %2"
                     :: "v"(lds_lab + off), "v"(off), "s"(glab) : "memory");
    }
    asm volatile("s_wait_asynccnt 0" ::: "memory");

    // Sum of raw outputs for this row (the "- outputs" term).
    // Stride-32-float LDS reads: lanes hit 32 distinct banks (64 banks x 4B) -> conflict free.
    float osum = 0.f;
#pragma unroll 4
    for (int k = lane; k < N_COLS; k += LANES) osum += s_out[k];

    // Pass 1: local inclusive logcumsumexp scans, 4 independent chains per lane for ILP.
    const int seg = lane * SEG;
    float acc0 = -INFINITY, acc1 = -INFINITY, acc2 = -INFINITY, acc3 = -INFINITY;
    for (int j = 0; j < CHLEN; ++j) {
        float g0 = s_out[s_lab[seg + 0 * CHLEN + j]];
        float g1 = s_out[s_lab[seg + 1 * CHLEN + j]];
        float g2 = s_out[s_lab[seg + 2 * CHLEN + j]];
        float g3 = s_out[s_lab[seg + 3 * CHLEN + j]];
        acc0 = lae(acc0, g0); Lbuf[seg + 0 * CHLEN + j] = acc0;
        acc1 = lae(acc1, g1); Lbuf[seg + 1 * CHLEN + j] = acc1;
        acc2 = lae(acc2, g2); Lbuf[seg + 2 * CHLEN + j] = acc2;
        acc3 = lae(acc3, g3); Lbuf[seg + 3 * CHLEN + j] = acc3;
    }

    // Lane total, then inclusive logaddexp scan across the wave (wave32 shuffles).
    float t = lae(lae(acc0, acc1), lae(acc2, acc3));
#pragma unroll
    for (int d = 1; d < 32; d <<= 1) {
        float o = __shfl_up(t, d, 32);
        if (lane >= d) t = lae(t, o);
    }
    float P = __shfl_up(t, 1, 32);
    if (lane == 0) P = -INFINITY;

    // Per-chain exclusive bases within the lane.
    float b0 = P;
    float b1 = lae(b0, acc0);
    float b2 = lae(b1, acc1);
    float b3 = lae(b2, acc2);

    // Pass 2: combine prefixes with local scans and accumulate sum of scores.
    float ssum = 0.f;
    for (int j = 0; j < CHLEN; ++j) {
        ssum += lae(b0, Lbuf[seg + 0 * CHLEN + j]);
        ssum += lae(b1, Lbuf[seg + 1 * CHLEN + j]);
        ssum += lae(b2, Lbuf[seg + 2 * CHLEN + j]);
        ssum += lae(b3, Lbuf[seg + 3 * CHLEN + j]);
    }

    // Row partial = sum(scores) - sum(outputs); deterministic wave tree-reduce.
    float v = ssum - osum;
#pragma unroll
    for (int d = 16; d > 0; d >>= 1) v += __shfl_down(v, d, 32);
    if (lane == 0) partial[row] = v;
}

__global__ __launch_bounds__(256) void listmle_reduce(
    const float* __restrict__ partial,
    float*       __restrict__ out)
{
    __shared__ double sm[256];
    double acc = 0.0;
    for (int i = threadIdx.x; i < B_ROWS; i += 256) acc += (double)partial[i];
    sm[threadIdx.x] = acc;
    __syncthreads();
    for (int s = 128; s > 0; s >>= 1) {
        if ((int)threadIdx.x < s) sm[threadIdx.x] += sm[threadIdx.x + s];
        __syncthreads();
    }
    if (threadIdx.x == 0)
        out[0] = (float)(sm[0] / ((double)B_ROWS * (double)N_COLS));
}

extern "C" void kernel_launch(void* const* d_in, const int* in_sizes, int n_in,
                              void* d_out, int out_size, void* d_ws, size_t ws_size,
                              hipStream_t stream) {
    const float* outputs = (const float*)d_in[0];
    const int*   labels  = (const int*)d_in[1];
    float*       partial = (float*)d_ws;   // B_ROWS floats = 32 KB of scratch

    listmle_rows<<<B_ROWS, 32, 0, stream>>>(outputs, labels, partial);
    listmle_reduce<<<1, 256, 0, stream>>>(partial, (float*)d_out);
}